// RGCN2_25168508354750
// MI455X (gfx1250) — compile-verified
//
#include <hip/hip_runtime.h>
#include <hip/hip_bf16.h>

#define RNUM 14
#define NN   100000
#define EE   1600000
#define BB   8
#define DD   64

typedef __attribute__((ext_vector_type(2))) float v2f;
typedef __attribute__((ext_vector_type(8))) float v8f;

// ---- ordered-uint encoding of float for integer atomic max ----
// enc preserves total order of floats; enc(-inf) = 0x007FFFFF
#define NEGINF_ENC 0x007FFFFFu
__device__ __forceinline__ unsigned enc_f(float f) {
    unsigned u = __float_as_uint(f);
    return (u & 0x80000000u) ? ~u : (u | 0x80000000u);
}
__device__ __forceinline__ float dec_f(unsigned u) {
    return (u & 0x80000000u) ? __uint_as_float(u & 0x7FFFFFFFu)
                             : __uint_as_float(~u);
}

// ---- expand block-diagonal layer-1 weights: (R,B,8,8) -> (R,64,64) ----
__global__ void rgcn_build_w1full(const float* __restrict__ w1,
                                  float* __restrict__ w1full) {
    int t = blockIdx.x * blockDim.x + threadIdx.x;            // r*4096 + i*64 + o
    if (t >= RNUM * DD * DD) return;
    int r = t / (DD * DD);
    int i = (t / DD) % DD;
    int o = t % DD;
    int bi = i >> 3, bo = o >> 3;
    float v = 0.f;
    if (bi == bo)
        v = w1[((r * BB + bi) * 8 + (i & 7)) * 8 + (o & 7)];
    w1full[t] = v;
}

// ---- basis combine layer-2 weights: W2[r] = sum_b comp2[r,b]*bases2[b] ----
__global__ void rgcn_build_w2(const float* __restrict__ comp2,
                              const float* __restrict__ bases2,
                              float* __restrict__ w2) {
    int t = blockIdx.x * blockDim.x + threadIdx.x;
    if (t >= RNUM * DD * DD) return;
    int r  = t / (DD * DD);
    int io = t % (DD * DD);
    float acc = 0.f;
#pragma unroll
    for (int b = 0; b < BB; ++b)
        acc += comp2[r * BB + b] * bases2[b * DD * DD + io];
    w2[t] = acc;
}

// ---- fp32 WMMA GEMM: D(nrows x 64) = X(nrows x 64) @ W(64 x 64) [+ bias] ----
// one wave32 computes one 16x16 output tile via 16x V_WMMA_F32_16X16X4_F32
__global__ void rgcn_gemm64_wmma(const float* __restrict__ X,
                                 const float* __restrict__ W,
                                 const float* __restrict__ bias,
                                 float* __restrict__ D,
                                 int nrows, int totalWaves) {
    int lane = threadIdx.x & 31;
    int wid  = blockIdx.x * (blockDim.x >> 5) + (threadIdx.x >> 5);
    if (wid >= totalWaves) return;                  // wave-uniform: EXEC all 1s at WMMA
    int tr = wid >> 2;                              // row tile (16 rows)
    int tc = wid & 3;                               // col tile (16 cols)
    int lo  = lane & 15;
    int hi  = lane >> 4;                            // 0: K=0,1 ; 1: K=2,3 within a x4 step
    int m   = tr * 16 + lo;                         // A row for this lane
    int col = tc * 16 + lo;                         // B/D column for this lane

    const float* xr = X + (size_t)m * DD + 2 * hi;  // pairs (k, k+1) at k = k0 + 2*hi
    v8f c = {};
#pragma unroll
    for (int k0 = 0; k0 < DD; k0 += 4) {
        v2f a = *(const v2f*)(xr + k0);             // X[m][k0+2hi], X[m][k0+2hi+1]
        int kb = k0 + 2 * hi;
        v2f b;
        b.x = W[kb * DD + col];
        b.y = W[(kb + 1) * DD + col];
        c = __builtin_amdgcn_wmma_f32_16x16x4_f32(
                false, a, false, b, (short)0, c, false, false);
    }
    float bv = bias ? bias[col] : 0.f;
#pragma unroll
    for (int v = 0; v < 8; ++v) {
        int row = tr * 16 + v + 8 * hi;             // C/D layout: VGPR v, halves +8
        D[(size_t)row * DD + col] = c[v] + bv;
    }
}

// ---- fill agg with enc(-inf) ----
__global__ void rgcn_init_agg(unsigned* __restrict__ agg, int n) {
    int t = blockIdx.x * blockDim.x + threadIdx.x;
    if (t < n) agg[t] = NEGINF_ENC;
}

// ---- scatter-max for edges of one relation: agg[dst,:] = max(H[src,:]) ----
// 4 threads per edge, 16 components each (4x float4 loads)
__global__ void rgcn_scatter_max(const int* __restrict__ src,
                                 const int* __restrict__ dst,
                                 const int* __restrict__ et,
                                 const float* __restrict__ H,
                                 unsigned* __restrict__ agg,
                                 int rel) {
    int t = blockIdx.x * blockDim.x + threadIdx.x;
    if (t >= 4 * EE) return;
    int e = t >> 2, part = t & 3;
    if (et[e] != rel) return;
    int s = src[e], d = dst[e];
    const float4* hp = (const float4*)(H + (size_t)s * DD + part * 16);
    unsigned* ap = agg + (size_t)d * DD + part * 16;
#pragma unroll
    for (int q = 0; q < 4; ++q) {
        float4 v = hp[q];
        atomicMax(ap + q * 4 + 0, enc_f(v.x));
        atomicMax(ap + q * 4 + 1, enc_f(v.y));
        atomicMax(ap + q * 4 + 2, enc_f(v.z));
        atomicMax(ap + q * 4 + 3, enc_f(v.w));
    }
}

// ---- acc += (agg empty ? 0 : decoded max) ----
__global__ void rgcn_accum(float* __restrict__ acc,
                           const unsigned* __restrict__ agg, int n) {
    int t = blockIdx.x * blockDim.x + threadIdx.x;
    if (t >= n) return;
    unsigned u = agg[t];
    if (u != NEGINF_ENC) acc[t] += dec_f(u);
}

// ---- in-place ReLU ----
__global__ void rgcn_relu(float* __restrict__ x, int n) {
    int t = blockIdx.x * blockDim.x + threadIdx.x;
    if (t < n) x[t] = fmaxf(x[t], 0.f);
}

extern "C" void kernel_launch(void* const* d_in, const int* in_sizes, int n_in,
                              void* d_out, int out_size, void* d_ws, size_t ws_size,
                              hipStream_t stream) {
    const float* x       = (const float*)d_in[0];
    const int*   eidx    = (const int*)d_in[1];   // (2,E): [0..E) src, [E..2E) dst
    const int*   etype   = (const int*)d_in[2];
    const float* w1      = (const float*)d_in[3];
    const float* root1   = (const float*)d_in[4];
    const float* bias1   = (const float*)d_in[5];
    const float* comp2   = (const float*)d_in[6];
    const float* bases2  = (const float*)d_in[7];
    const float* root2   = (const float*)d_in[8];
    const float* bias2   = (const float*)d_in[9];
    float* out = (float*)d_out;

    const int* src = eidx;
    const int* dst = eidx + EE;

    const size_t N64 = (size_t)NN * DD;           // 6.4M elems
    char* p = (char*)d_ws;
    float*    H    = (float*)p;    p += N64 * sizeof(float);
    unsigned* agg  = (unsigned*)p; p += N64 * sizeof(unsigned);
    float*    x1   = (float*)p;    p += N64 * sizeof(float);
    float*    w1f  = (float*)p;    p += (size_t)RNUM * DD * DD * sizeof(float);
    float*    w2   = (float*)p;    p += (size_t)RNUM * DD * DD * sizeof(float);

    const int wBlk = 57344;                                   // R*64*64
    rgcn_build_w1full<<<(wBlk + 255) / 256, 256, 0, stream>>>(w1, w1f);
    rgcn_build_w2   <<<(wBlk + 255) / 256, 256, 0, stream>>>(comp2, bases2, w2);

    const int totalWaves = (NN / 16) * 4;                     // 25000 tiles
    const int gemmBlocks = (totalWaves + 7) / 8;              // 8 waves / 256-thr block
    const int elemBlocks = (int)((N64 + 255) / 256);
    const int edgeBlocks = (4 * EE + 255) / 256;

    // ================= Layer 1 =================
    // acc (=x1) = x @ root1 + bias1
    rgcn_gemm64_wmma<<<gemmBlocks, 256, 0, stream>>>(x, root1, bias1, x1, NN, totalWaves);
    for (int r = 0; r < RNUM; ++r) {
        rgcn_gemm64_wmma<<<gemmBlocks, 256, 0, stream>>>(x, w1f + (size_t)r * DD * DD,
                                                         nullptr, H, NN, totalWaves);
        rgcn_init_agg   <<<elemBlocks, 256, 0, stream>>>(agg, (int)N64);
        rgcn_scatter_max<<<edgeBlocks, 256, 0, stream>>>(src, dst, etype, H, agg, r);
        rgcn_accum      <<<elemBlocks, 256, 0, stream>>>(x1, agg, (int)N64);
    }
    rgcn_relu<<<elemBlocks, 256, 0, stream>>>(x1, (int)N64);

    // ================= Layer 2 =================
    // acc (=out) = x1 @ root2 + bias2
    rgcn_gemm64_wmma<<<gemmBlocks, 256, 0, stream>>>(x1, root2, bias2, out, NN, totalWaves);
    for (int r = 0; r < RNUM; ++r) {
        rgcn_gemm64_wmma<<<gemmBlocks, 256, 0, stream>>>(x1, w2 + (size_t)r * DD * DD,
                                                         nullptr, H, NN, totalWaves);
        rgcn_init_agg   <<<elemBlocks, 256, 0, stream>>>(agg, (int)N64);
        rgcn_scatter_max<<<edgeBlocks, 256, 0, stream>>>(src, dst, etype, H, agg, r);
        rgcn_accum      <<<elemBlocks, 256, 0, stream>>>(out, agg, (int)N64);
    }
}